// Attention_51977694216270
// MI455X (gfx1250) — compile-verified
//
#include <hip/hip_runtime.h>
#include <hip/hip_bf16.h>

typedef __attribute__((ext_vector_type(2))) float v2f;
typedef __attribute__((ext_vector_type(8))) float v8f;

#define NB   8
#define SRC  512
#define TGT  128
#define DIM  512

// ---------------- fast tanh (prefer gfx1250 v_tanh_f32) ----------------
__device__ __forceinline__ float fast_tanh(float x) {
#if __has_builtin(__builtin_amdgcn_tanhf)
    return __builtin_amdgcn_tanhf(x);
#else
    // tanh(x) = 1 - 2/(exp(2x)+1); exp(2x) = exp2(x * 2*log2(e))
    float e2 = __builtin_amdgcn_exp2f(x * 2.885390081777927f);
    return 1.0f - 2.0f * __builtin_amdgcn_rcpf(e2 + 1.0f);
#endif
}

// ---------------- f32 WMMA GEMM: C(MxN) = X(MxK) * W(KxN), row-major ----
// Block: 256 threads = 8 waves (2x4), wave computes 32x32 (2x2 tiles of 16x16).
// Block tile: 64x128, K staged through LDS in chunks of 32.
// Tiles are stored K-PAIR-major (float2 of K,K+1 per element) so each WMMA
// fragment (lane p=l>>4, q=l&15 needs elements K=2p and 2p+1) is one aligned
// ds_load_b64 straight into an even VGPR pair -- no packing movs.
#define BM 64
#define BN 128
#define KC 32

__global__ __launch_bounds__(256)
void gemm_f32_wmma(const float* __restrict__ Xb, const float* __restrict__ Wb,
                   float* __restrict__ Cb, int M, int N, int K,
                   long long sx, long long sw, long long sc)
{
    __shared__ float2 As2[KC / 2][BM];   // As2[k/2][m] = (X[m][k], X[m][k+1])   8 KB
    __shared__ float2 Bs2[KC / 2][BN];   // Bs2[k/2][n] = (W[k][n], W[k+1][n])  16 KB

    const int tid  = threadIdx.x;
    const int bmi  = blockIdx.x;
    const int bni  = blockIdx.y;
    const int bz   = blockIdx.z;

    const float* X = Xb + (long long)bz * sx;
    const float* W = Wb + (long long)bz * sw;
    float*       C = Cb + (long long)bz * sc;

    const int w    = tid >> 5;
    const int lane = tid & 31;
    const int wm   = w >> 2;        // 0..1
    const int wn   = w & 3;         // 0..3
    const int p    = lane >> 4;     // K-pair selector
    const int q    = lane & 15;     // M (A) / N (B,C,D) index

    v8f acc[2][2] = {};             // zero accumulators

    // loader coords
    const int ra = tid >> 2, kq = tid & 3;    // A: row 0..63, k-quarter 0..3
    const int kp = tid >> 4, ng = tid & 15;   // B: k-pair 0..15, n-group 0..15

    const long long rowA  = (long long)(bmi * BM + ra) * K;
    const long long rowB0 = (long long)(2 * kp) * N;
    const int colB = bni * BN + ng * 8;

    for (int kb = 0; kb < K; kb += KC) {
        // ---- stage A: transpose to K-pair-major ----
        #pragma unroll
        for (int pass = 0; pass < 2; ++pass) {
            const int kk = kq * 8 + pass * 4;          // even
            float4 va = *(const float4*)(X + rowA + kb + kk);
            As2[(kk >> 1) + 0][ra] = make_float2(va.x, va.y);
            As2[(kk >> 1) + 1][ra] = make_float2(va.z, va.w);
        }
        // ---- stage B: interleave two consecutive K rows ----
        #pragma unroll
        for (int u = 0; u < 2; ++u) {
            float4 r0 = *(const float4*)(W + (long long)kb * N + rowB0 + colB + u * 4);
            float4 r1 = *(const float4*)(W + (long long)kb * N + rowB0 + N + colB + u * 4);
            Bs2[kp][ng * 8 + u * 4 + 0] = make_float2(r0.x, r1.x);
            Bs2[kp][ng * 8 + u * 4 + 1] = make_float2(r0.y, r1.y);
            Bs2[kp][ng * 8 + u * 4 + 2] = make_float2(r0.z, r1.z);
            Bs2[kp][ng * 8 + u * 4 + 3] = make_float2(r0.w, r1.w);
        }
        __syncthreads();

        #pragma unroll
        for (int kk = 0; kk < KC; kk += 4) {
            const int kpair = (kk >> 1) + p;           // K = 2*kpair, 2*kpair+1
            float2 fa0 = As2[kpair][wm * 32 + q];
            float2 fa1 = As2[kpair][wm * 32 + 16 + q];
            float2 fb0 = Bs2[kpair][wn * 32 + q];
            float2 fb1 = Bs2[kpair][wn * 32 + 16 + q];
            v2f a0; a0.x = fa0.x; a0.y = fa0.y;
            v2f a1; a1.x = fa1.x; a1.y = fa1.y;
            v2f b0; b0.x = fb0.x; b0.y = fb0.y;
            v2f b1; b1.x = fb1.x; b1.y = fb1.y;

            acc[0][0] = __builtin_amdgcn_wmma_f32_16x16x4_f32(false, a0, false, b0, (short)0, acc[0][0], false, false);
            acc[0][1] = __builtin_amdgcn_wmma_f32_16x16x4_f32(false, a0, false, b1, (short)0, acc[0][1], false, false);
            acc[1][0] = __builtin_amdgcn_wmma_f32_16x16x4_f32(false, a1, false, b0, (short)0, acc[1][0], false, false);
            acc[1][1] = __builtin_amdgcn_wmma_f32_16x16x4_f32(false, a1, false, b1, (short)0, acc[1][1], false, false);
        }
        __syncthreads();
    }

    // ---- epilogue: C/D layout row = j + 8*p, col = q ----
    #pragma unroll
    for (int tm = 0; tm < 2; ++tm)
        #pragma unroll
        for (int tn = 0; tn < 2; ++tn) {
            const int col = bni * BN + wn * 32 + tn * 16 + q;
            #pragma unroll
            for (int j = 0; j < 8; ++j) {
                const int row = bmi * BM + wm * 32 + tm * 16 + 8 * p + j;
                C[(long long)row * N + col] = acc[tm][tn][j];
            }
        }
}

// ---------------- scores: e[b,t,s] = sum_k Va[k]*tanh(dp[t,k]+mp[s,k]) ----
// One block handles 8 decoder rows (t) of one batch: dp rows + Va in LDS
// (broadcast reads), each lane owns one s and streams mp[s,:] as float4.
__global__ __launch_bounds__(256)
void scores_tanh(const float* __restrict__ dp, const float* __restrict__ mp,
                 const float* __restrict__ Va, float* __restrict__ e)
{
    __shared__ float dps[8 * DIM];   // 16 KB
    __shared__ float vas[DIM];       // 2 KB

    const int tid = threadIdx.x;
    const int blk = blockIdx.x;      // 0..127
    const int b   = blk >> 4;        // 16 blocks per batch (TGT/8)
    const int t0  = (blk & 15) * 8;

    const float* dpb = dp + (long long)(b * TGT + t0) * DIM;
    for (int i = tid; i < (8 * DIM) / 4; i += 256)
        ((float4*)dps)[i] = ((const float4*)dpb)[i];
    for (int i = tid; i < DIM / 4; i += 256)
        ((float4*)vas)[i] = ((const float4*)Va)[i];
    __syncthreads();

    const int w = tid >> 5, lane = tid & 31;
    const int t = t0 + w;
    const float* mpb  = mp + (long long)b * SRC * DIM;
    const float* drow = dps + w * DIM;
    float* erow = e + (long long)(b * TGT + t) * SRC;

    for (int s0 = 0; s0 < SRC; s0 += 32) {
        const int s = s0 + lane;
        const float* mrow = mpb + (long long)s * DIM;
        if (s0 + 32 < SRC) __builtin_prefetch(mrow + 32 * DIM, 0, 0);
        float acc = 0.0f;
        #pragma unroll 4
        for (int k = 0; k < DIM; k += 4) {
            float4 m4 = *(const float4*)(mrow + k);
            float4 d4 = *(const float4*)(drow + k);
            float4 v4 = *(const float4*)(vas + k);
            acc += v4.x * fast_tanh(d4.x + m4.x);
            acc += v4.y * fast_tanh(d4.y + m4.y);
            acc += v4.z * fast_tanh(d4.z + m4.z);
            acc += v4.w * fast_tanh(d4.w + m4.w);
        }
        erow[s] = acc;
    }
}

// ---------------- row softmax over s (length 512), mask applied ----------
__global__ __launch_bounds__(256)
void softmax_rows(float* __restrict__ e, const unsigned char* __restrict__ mask)
{
    const int r   = blockIdx.x;          // 0..B*TGT-1
    const int b   = r >> 7;              // TGT = 128 rows per batch
    float* row = e + (long long)r * SRC;
    const unsigned char* mrow = mask + (long long)b * SRC;

    const int tid = threadIdx.x, w = tid >> 5, lane = tid & 31;
    float x0 = row[tid];
    float x1 = row[tid + 256];
    if (!mrow[tid])       x0 = -INFINITY;
    if (!mrow[tid + 256]) x1 = -INFINITY;

    __shared__ float redm[8];
    __shared__ float reds[8];

    float v = fmaxf(x0, x1);
    for (int o = 16; o; o >>= 1) v = fmaxf(v, __shfl_xor(v, o, 32));
    if (lane == 0) redm[w] = v;
    __syncthreads();
    float m = redm[0];
    #pragma unroll
    for (int i = 1; i < 8; ++i) m = fmaxf(m, redm[i]);

    const float L2E = 1.4426950408889634f;
    float p0 = __builtin_amdgcn_exp2f((x0 - m) * L2E);
    float p1 = __builtin_amdgcn_exp2f((x1 - m) * L2E);

    float s = p0 + p1;
    for (int o = 16; o; o >>= 1) s += __shfl_xor(s, o, 32);
    if (lane == 0) reds[w] = s;
    __syncthreads();
    float tot = 0.0f;
    #pragma unroll
    for (int i = 0; i < 8; ++i) tot += reds[i];

    float inv = 1.0f / tot;
    row[tid]       = p0 * inv;
    row[tid + 256] = p1 * inv;
}

// ---------------- launch ----------------
extern "C" void kernel_launch(void* const* d_in, const int* in_sizes, int n_in,
                              void* d_out, int out_size, void* d_ws, size_t ws_size,
                              hipStream_t stream)
{
    (void)in_sizes; (void)n_in; (void)out_size; (void)ws_size;

    const float*         memory = (const float*)d_in[0];   // (8,512,512)
    const float*         dec    = (const float*)d_in[1];   // (8,128,512)
    const unsigned char* mask   = (const unsigned char*)d_in[2]; // (8,512) bool
    const float*         Wa     = (const float*)d_in[3];   // (1024,512)
    const float*         Va     = (const float*)d_in[4];   // (512,)
    float* out = (float*)d_out;                            // (8,128,512)

    float* dp = (float*)d_ws;                              // 1024x512  (2 MB)
    float* mp = dp + (long long)NB * TGT * DIM;            // 4096x512  (8 MB)
    float* ea = mp + (long long)NB * SRC * DIM;            // 1024x512  (2 MB)

    dim3 blk(256);

    // dp = decoder_state(1024x512) @ Wa_d(512x512)
    gemm_f32_wmma<<<dim3((NB * TGT) / BM, DIM / BN, 1), blk, 0, stream>>>(
        dec, Wa, dp, NB * TGT, DIM, DIM, 0, 0, 0);

    // mp = memory(4096x512) @ Wa_m(512x512)
    gemm_f32_wmma<<<dim3((NB * SRC) / BM, DIM / BN, 1), blk, 0, stream>>>(
        memory, Wa + (long long)DIM * DIM, mp, NB * SRC, DIM, DIM, 0, 0, 0);

    // e[b,t,s]
    scores_tanh<<<dim3((NB * TGT) / 8), blk, 0, stream>>>(dp, mp, Va, ea);

    // softmax rows -> attention weights (in place)
    softmax_rows<<<dim3(NB * TGT), blk, 0, stream>>>(ea, mask);

    // context[b] = a[b](128x512) @ memory[b](512x512), batched over b
    gemm_f32_wmma<<<dim3(TGT / BM, DIM / BN, NB), blk, 0, stream>>>(
        ea, memory, out, TGT, DIM, DIM,
        (long long)TGT * SRC, (long long)SRC * DIM, (long long)TGT * DIM);
}